// CodeBookDecoder_20409684591390
// MI455X (gfx1250) — compile-verified
//
#include <hip/hip_runtime.h>
#include <hip/hip_bf16.h>
#include <math.h>

typedef float v2f __attribute__((ext_vector_type(2)));
typedef float v8f __attribute__((ext_vector_type(8)));

#define DEVFN static __device__ __forceinline__

namespace cfg {
constexpr int B = 256, C = 64, L = 512, NLAB = 10, EMB = 256, NFREQ = 9;
constexpr int CIN = C + 2 * NFREQ;            // 82
constexpr int FIN = 4 * C, FH = C, FOUT = C;  // 256, 64, 64
constexpr int P = FIN * FH + FH + FH * FOUT + FOUT + FIN * FOUT + FOUT;  // 37056
constexpr int O_WH = 0;
constexpr int O_BH = FIN * FH;            // 16384
constexpr int O_WL = O_BH + FH;           // 16448
constexpr int O_BL = O_WL + FH * FOUT;    // 20544
constexpr int O_WS = O_BL + FOUT;         // 20608
constexpr int O_BS = O_WS + FIN * FOUT;   // 36992
constexpr int XNS = 528;                  // LDS row stride (floats) for B-operand tiles
}

DEVFN float lrelu(float x) { return x > 0.0f ? x : 0.2f * x; }

// D = A(16x4 f32) * B(4x16 f32) + C, fp32 accumulate.
DEVFN v8f wmma4(v2f a, v2f b, v8f c) {
  return __builtin_amdgcn_wmma_f32_16x16x4_f32(false, a, false, b, (short)0, c,
                                               false, false);
}

// ---------------------------------------------------------------------------
// K1: yemb[b] = normalize(fc_w[:, y[b]] + fc_b)
// ---------------------------------------------------------------------------
__global__ __launch_bounds__(256) void yemb_k(const float* __restrict__ fc_w,
                                              const float* __restrict__ fc_b,
                                              const int* __restrict__ y,
                                              float* __restrict__ yemb) {
  using namespace cfg;
  __shared__ float red[256];
  int b = blockIdx.x, e = threadIdx.x;
  int lab = y[b];
  float v = fc_w[e * NLAB + lab] + fc_b[e];
  red[e] = v * v;
  __syncthreads();
  for (int s = 128; s > 0; s >>= 1) {
    if (e < s) red[e] += red[e + s];
    __syncthreads();
  }
  float nrm = fmaxf(sqrtf(red[0]), 1e-12f);
  yemb[b * EMB + e] = v / nrm;
}

// ---------------------------------------------------------------------------
// K2: p[b, j] = sum_e yemb[b,e] * hyper_w[j,e] + hyper_b[j]  (WMMA f32)
// A = hyper_w (M=j, K=e, row-major), B[k,n] = yemb[n,k].
// grid = (2316/4, 16), block = 128 (4 waves, one 16x16 tile each).
// ---------------------------------------------------------------------------
__global__ __launch_bounds__(128) void hyper_k(const float* __restrict__ yemb,
                                               const float* __restrict__ hw,
                                               const float* __restrict__ hb,
                                               float* __restrict__ p) {
  using namespace cfg;
  int lane = threadIdx.x & 31, w = threadIdx.x >> 5;
  int m_tile = blockIdx.x * 4 + w;
  int n_tile = blockIdx.y;
  int hm = lane >> 4, l15 = lane & 15;
  int m = m_tile * 16 + l15;  // A row for this lane
  int n = n_tile * 16 + l15;  // B col for this lane
  v8f acc;
#pragma unroll
  for (int r = 0; r < 8; ++r) acc[r] = 0.0f;
  for (int kb = 0; kb < EMB / 4; ++kb) {
    int kk = kb * 4 + hm * 2;
    v2f a = *(const v2f*)(hw + (size_t)m * EMB + kk);
    v2f bb = *(const v2f*)(yemb + n * EMB + kk);
    acc = wmma4(a, bb, acc);
  }
#pragma unroll
  for (int r = 0; r < 8; ++r) {
    int mm = m_tile * 16 + r + 8 * hm;
    int nn = n_tile * 16 + l15;
    p[(size_t)nn * P + mm] = acc[r] + hb[mm];
  }
}

// ---------------------------------------------------------------------------
// K3: input res1d.  xin = concat(codes[b], pos_enc) (82 x 512) in LDS.
// xs = in_ws(1x1) @ xin ; dx = conv3(lrelu(xin), w0)+b0 ; dx = conv3(lrelu(dx), w1)+b1
// x = xs + 0.1 dx  -> global
// ---------------------------------------------------------------------------
__global__ __launch_bounds__(256) void pre_k(const float* __restrict__ codes,
                                             const float* __restrict__ w0,
                                             const float* __restrict__ b0,
                                             const float* __restrict__ w1,
                                             const float* __restrict__ b1,
                                             const float* __restrict__ wsw,
                                             float* __restrict__ xout) {
  using namespace cfg;
  __shared__ float xin[CIN * L];  // 82*512*4 = 168KB
  __shared__ float dxb[C * L];    // 64*512*4 = 128KB
  int b = blockIdx.x, tid = threadIdx.x;
  const float PI = 3.14159265358979323846f;
  for (int idx = tid; idx < CIN * L; idx += 256) {
    int r = idx >> 9, s = idx & (L - 1);
    float v;
    if (r < C) {
      v = codes[((size_t)b * C + r) * L + s];
    } else {
      int i = r - C;
      float f = (i < NFREQ) ? PI * exp2f(-(float)i) : PI * exp2f(-(float)(i - NFREQ));
      v = (i < NFREQ) ? sinf(f * (float)s) : cosf(f * (float)s);
    }
    xin[idx] = v;
  }
  __syncthreads();
  int c = tid >> 2, s0 = (tid & 3) * 128;
  // conv1: dxb = conv3(lrelu(xin), w0) + b0
  for (int sc = 0; sc < 4; ++sc) {
    int sb = s0 + sc * 32;
    float acc[32];
    float bias = b0[c];
#pragma unroll
    for (int j = 0; j < 32; ++j) acc[j] = bias;
    for (int i = 0; i < CIN; ++i) {
      const float* row = xin + i * L;
      for (int d = 0; d < 3; ++d) {
        float wv = w0[(c * CIN + i) * 3 + d];
#pragma unroll
        for (int j = 0; j < 32; ++j) {
          int ss = sb + j + d - 1;
          float xv = ((unsigned)ss < (unsigned)L) ? row[ss] : 0.0f;
          acc[j] = fmaf(wv, lrelu(xv), acc[j]);
        }
      }
    }
#pragma unroll
    for (int j = 0; j < 32; ++j) dxb[c * L + sb + j] = acc[j];
  }
  __syncthreads();
  // xs (1x1) + conv2; write x to global
  for (int sc = 0; sc < 4; ++sc) {
    int sb = s0 + sc * 32;
    float acc[32], xsv[32];
    float bias = b1[c];
#pragma unroll
    for (int j = 0; j < 32; ++j) { acc[j] = bias; xsv[j] = 0.0f; }
    for (int i = 0; i < CIN; ++i) {
      float wv = wsw[c * CIN + i];
      const float* row = xin + i * L;
#pragma unroll
      for (int j = 0; j < 32; ++j) xsv[j] = fmaf(wv, row[sb + j], xsv[j]);
    }
    for (int i = 0; i < C; ++i) {
      const float* row = dxb + i * L;
      for (int d = 0; d < 3; ++d) {
        float wv = w1[(c * C + i) * 3 + d];
#pragma unroll
        for (int j = 0; j < 32; ++j) {
          int ss = sb + j + d - 1;
          float xv = ((unsigned)ss < (unsigned)L) ? row[ss] : 0.0f;
          acc[j] = fmaf(wv, lrelu(xv), acc[j]);
        }
      }
    }
#pragma unroll
    for (int j = 0; j < 32; ++j)
      xout[((size_t)b * C + c) * L + sb + j] = xsv[j] + 0.1f * acc[j];
  }
}

// ---------------------------------------------------------------------------
// K4: fused 8-iteration hypernetwork loop. One block (1024 thr, 32 waves) per
// batch element; x stays in LDS across all iterations; dyna GEMMs via WMMA.
// ---------------------------------------------------------------------------
__global__ __launch_bounds__(1024) void iter_k(const float* __restrict__ pbuf,
                                               const float* __restrict__ leak_p,
                                               float* __restrict__ xg) {
  using namespace cfg;
  __shared__ float xl[C * L];     // raw x, 128KB
  __shared__ float xn[C * XNS];   // group tile / hidden acts, 132KB
  __shared__ float meanv[C], rstdv[C];
  int b = blockIdx.x, tid = threadIdx.x;
  int lane = tid & 31, w = tid >> 5;  // 32 waves
  const float* p = pbuf + (size_t)b * P;
  float leak = fminf(fmaxf(leak_p[0], 0.001f), 1000.0f);

  for (int idx = tid; idx < C * L; idx += 1024)
    xl[idx] = xg[(size_t)b * C * L + idx];
  __syncthreads();

  int m_tile = w & 3, n_group = w >> 2;  // 4 M-tiles x 8 N-groups (4 tiles each)
  int hm = lane >> 4, l15 = lane & 15;
  int mrow = m_tile * 16 + l15;  // A-operand row
  const float sker[3] = {1.0f, 2.0f, 1.0f};
  const float dker[3] = {-1.0f, 0.0f, 1.0f};

#pragma clang loop unroll(disable)
  for (int it = 0; it < 8; ++it) {
    // ---- instance norm statistics: 2 channels per wave ----
    for (int cc = 0; cc < 2; ++cc) {
      int c = w * 2 + cc;
      float s = 0.0f, sq = 0.0f;
      for (int j = lane; j < L; j += 32) {
        float v = xl[c * L + j];
        s += v;
        sq += v * v;
      }
#pragma unroll
      for (int off = 16; off > 0; off >>= 1) {
        s += __shfl_xor(s, off, 32);
        sq += __shfl_xor(sq, off, 32);
      }
      if (lane == 0) {
        float m = s * (1.0f / 512.0f);
        float var = sq * (1.0f / 512.0f) - m * m;
        meanv[c] = m;
        rstdv[c] = rsqrtf(var + 1e-5f);
      }
    }
    __syncthreads();

    v8f axs[4], adx[4];
#pragma unroll
    for (int t = 0; t < 4; ++t)
#pragma unroll
      for (int r = 0; r < 8; ++r) { axs[t][r] = 0.0f; adx[t][r] = 0.0f; }

#pragma unroll
    for (int g = 0; g < 4; ++g) {
      // ---- generate group g of xn (normalized / sin(sobel)) ----
      {
        int c = tid >> 4;            // 0..63
        int p0 = (tid & 15) * 32;    // 32 positions each
        float m = meanv[c], r = rstdv[c];
        if (g == 0) {
          for (int j = 0; j < 32; ++j) {
            int s = p0 + j;
            xn[c * XNS + s] = (xl[c * L + s] - m) * r;
          }
        } else {
          int axis = g - 1;
          for (int j = 0; j < 32; ++j) {
            int s = p0 + j;
            int z = s >> 6, y = (s >> 3) & 7, x = s & 7;
            float acc = 0.0f;
#pragma unroll
            for (int dz = -1; dz <= 1; ++dz) {
              int zz = z + dz;
              if ((unsigned)zz >= 8u) continue;
              float wz = (axis == 0) ? dker[dz + 1] : sker[dz + 1];
#pragma unroll
              for (int dy = -1; dy <= 1; ++dy) {
                int yy = y + dy;
                if ((unsigned)yy >= 8u) continue;
                float wy = (axis == 1) ? dker[dy + 1] : sker[dy + 1];
#pragma unroll
                for (int dx = -1; dx <= 1; ++dx) {
                  int xx = x + dx;
                  if ((unsigned)xx >= 8u) continue;
                  float wx = (axis == 2) ? dker[dx + 1] : sker[dx + 1];
                  float wgt = wz * wy * wx;
                  float v = (xl[c * L + (zz << 6) + (yy << 3) + xx] - m) * r;
                  acc = fmaf(wgt, v, acc);
                }
              }
            }
            xn[c * XNS + s] = sinf(acc);
          }
        }
      }
      __syncthreads();
      // ---- WMMA accumulate: xs += ws_g * xn_g ; dxh += wh_g * xn_g ----
      const float* pws = p + O_WS;
      const float* pwh = p + O_WH;
      for (int kb = 0; kb < 16; ++kb) {
        int kk = kb * 4 + hm * 2;       // K within group (per lane-half)
        int kglob = g * 64 + kk;        // K within FIN
        v2f aws = *(const v2f*)(pws + mrow * FIN + kglob);
        v2f awh = *(const v2f*)(pwh + mrow * FIN + kglob);
#pragma unroll
        for (int t = 0; t < 4; ++t) {
          int n0 = (n_group * 4 + t) * 16 + l15;
          v2f bb;
          bb.x = xn[kk * XNS + n0];
          bb.y = xn[(kk + 1) * XNS + n0];
          axs[t] = wmma4(aws, bb, axs[t]);
          adx[t] = wmma4(awh, bb, adx[t]);
        }
      }
      __syncthreads();
    }

    // ---- hidden = lrelu(dxh + bh) -> LDS (reuse xn) ----
#pragma unroll
    for (int r = 0; r < 8; ++r) {
      int mm = m_tile * 16 + r + 8 * hm;
      float bh = p[O_BH + mm];
#pragma unroll
      for (int t = 0; t < 4; ++t) {
        int nn = (n_group * 4 + t) * 16 + l15;
        xn[mm * XNS + nn] = lrelu(adx[t][r] + bh);
      }
    }
    __syncthreads();

    // ---- dx2 = wl * hidden (WMMA, K=64) ----
#pragma unroll
    for (int t = 0; t < 4; ++t)
#pragma unroll
      for (int r = 0; r < 8; ++r) adx[t][r] = 0.0f;
    const float* pwl = p + O_WL;
    for (int kb = 0; kb < 16; ++kb) {
      int kk = kb * 4 + hm * 2;
      v2f awl = *(const v2f*)(pwl + mrow * FH + kk);
#pragma unroll
      for (int t = 0; t < 4; ++t) {
        int n0 = (n_group * 4 + t) * 16 + l15;
        v2f bb;
        bb.x = xn[kk * XNS + n0];
        bb.y = xn[(kk + 1) * XNS + n0];
        adx[t] = wmma4(awl, bb, adx[t]);
      }
    }

    // ---- residual update: x += leak * (xs + bs + 0.1*(dx2 + bl)) ----
#pragma unroll
    for (int r = 0; r < 8; ++r) {
      int mm = m_tile * 16 + r + 8 * hm;
      float bs = p[O_BS + mm];
      float bl = p[O_BL + mm];
#pragma unroll
      for (int t = 0; t < 4; ++t) {
        int nn = (n_group * 4 + t) * 16 + l15;
        xl[mm * L + nn] +=
            leak * (axs[t][r] + bs + 0.1f * (adx[t][r] + bl));
      }
    }
    __syncthreads();
  }

  for (int idx = tid; idx < C * L; idx += 1024)
    xg[(size_t)b * C * L + idx] = xl[idx];
}

// ---------------------------------------------------------------------------
// K5: output res1d + cross-entropy partials + lat head.  One block per batch.
// ---------------------------------------------------------------------------
__global__ __launch_bounds__(256) void post_k(
    const float* __restrict__ xg, const float* __restrict__ codes,
    const float* __restrict__ ow0, const float* __restrict__ ob0,
    const float* __restrict__ ow1, const float* __restrict__ ob1,
    const float* __restrict__ l1w0, const float* __restrict__ l1b0,
    const float* __restrict__ l1w1, const float* __restrict__ l1b1,
    const float* __restrict__ l2w0, const float* __restrict__ l2b0,
    const float* __restrict__ l2w1, const float* __restrict__ l2b1,
    const float* __restrict__ l2ws, float* __restrict__ out_lat,
    float* __restrict__ loss_part) {
  using namespace cfg;
  __shared__ float xf[C * L];
  __shared__ float dxb[C * L];
  __shared__ float red[256];
  int b = blockIdx.x, tid = threadIdx.x;
  for (int idx = tid; idx < C * L; idx += 256)
    xf[idx] = xg[(size_t)b * C * L + idx];
  __syncthreads();
  int c = tid >> 2, s0 = (tid & 3) * 128;
  // out-res1d conv1
  for (int sc = 0; sc < 4; ++sc) {
    int sb = s0 + sc * 32;
    float acc[32];
    float bias = ob0[c];
#pragma unroll
    for (int j = 0; j < 32; ++j) acc[j] = bias;
    for (int i = 0; i < C; ++i) {
      const float* row = xf + i * L;
      for (int d = 0; d < 3; ++d) {
        float wv = ow0[(c * C + i) * 3 + d];
#pragma unroll
        for (int j = 0; j < 32; ++j) {
          int ss = sb + j + d - 1;
          float xv = ((unsigned)ss < (unsigned)L) ? row[ss] : 0.0f;
          acc[j] = fmaf(wv, lrelu(xv), acc[j]);
        }
      }
    }
#pragma unroll
    for (int j = 0; j < 32; ++j) dxb[c * L + sb + j] = acc[j];
  }
  __syncthreads();
  // out-res1d conv2 + residual (in place, identity shortcut)
  for (int sc = 0; sc < 4; ++sc) {
    int sb = s0 + sc * 32;
    float acc[32];
    float bias = ob1[c];
#pragma unroll
    for (int j = 0; j < 32; ++j) acc[j] = bias;
    for (int i = 0; i < C; ++i) {
      const float* row = dxb + i * L;
      for (int d = 0; d < 3; ++d) {
        float wv = ow1[(c * C + i) * 3 + d];
#pragma unroll
        for (int j = 0; j < 32; ++j) {
          int ss = sb + j + d - 1;
          float xv = ((unsigned)ss < (unsigned)L) ? row[ss] : 0.0f;
          acc[j] = fmaf(wv, lrelu(xv), acc[j]);
        }
      }
    }
#pragma unroll
    for (int j = 0; j < 32; ++j) xf[c * L + sb + j] += 0.1f * acc[j];
  }
  __syncthreads();
  // cross-entropy partial (deterministic block reduction)
  float lsum = 0.0f;
  for (int l = tid; l < L; l += 256) {
    float mx = -1e30f;
    for (int c2 = 0; c2 < C; ++c2) mx = fmaxf(mx, xf[c2 * L + l]);
    float se = 0.0f;
    for (int c2 = 0; c2 < C; ++c2) se += expf(xf[c2 * L + l] - mx);
    float lse = logf(se) + mx;
    int idx = 0;
    float bv = codes[(size_t)b * C * L + l];
    for (int c2 = 1; c2 < C; ++c2) {
      float v = codes[((size_t)b * C + c2) * L + l];
      if (v > bv) { bv = v; idx = c2; }
    }
    lsum += lse - xf[idx * L + l];
  }
  red[tid] = lsum;
  __syncthreads();
  for (int s = 128; s > 0; s >>= 1) {
    if (tid < s) red[tid] += red[tid + s];
    __syncthreads();
  }
  if (tid == 0) loss_part[b] = red[0];
  __syncthreads();
  // lat res1d (width-1): dxA = l1w0 @ lrelu(xf) + l1b0
  for (int sc = 0; sc < 4; ++sc) {
    int sb = s0 + sc * 32;
    float acc[32];
    float bias = l1b0[c];
#pragma unroll
    for (int j = 0; j < 32; ++j) acc[j] = bias;
    for (int i = 0; i < C; ++i) {
      float wv = l1w0[c * C + i];
      const float* row = xf + i * L;
#pragma unroll
      for (int j = 0; j < 32; ++j) acc[j] = fmaf(wv, lrelu(row[sb + j]), acc[j]);
    }
#pragma unroll
    for (int j = 0; j < 32; ++j) dxb[c * L + sb + j] = acc[j];
  }
  __syncthreads();
  // lat = xf + 0.1*(l1w1 @ lrelu(dxA) + l1b1)  (in place)
  for (int sc = 0; sc < 4; ++sc) {
    int sb = s0 + sc * 32;
    float acc[32];
    float bias = l1b1[c];
#pragma unroll
    for (int j = 0; j < 32; ++j) acc[j] = bias;
    for (int i = 0; i < C; ++i) {
      float wv = l1w1[c * C + i];
      const float* row = dxb + i * L;
#pragma unroll
      for (int j = 0; j < 32; ++j) acc[j] = fmaf(wv, lrelu(row[sb + j]), acc[j]);
    }
#pragma unroll
    for (int j = 0; j < 32; ++j) xf[c * L + sb + j] += 0.1f * acc[j];
  }
  __syncthreads();
  // l2 head -> out_lat
  for (int l = tid; l < L; l += 256) {
    float xs = 0.0f, dh = l2b0[0];
    for (int i = 0; i < C; ++i) {
      float v = xf[i * L + l];
      xs = fmaf(l2ws[i], v, xs);
      dh = fmaf(l2w0[i], lrelu(v), dh);
    }
    float dxo = l2b1[0] + l2w1[0] * lrelu(dh);
    out_lat[(size_t)b * L + l] = xs + 0.1f * dxo;
  }
}

// ---------------------------------------------------------------------------
// K6: final deterministic loss reduction
// ---------------------------------------------------------------------------
__global__ __launch_bounds__(256) void loss_k(const float* __restrict__ lp,
                                              float* __restrict__ out) {
  using namespace cfg;
  __shared__ float red[256];
  int t = threadIdx.x;
  red[t] = lp[t];
  __syncthreads();
  for (int s = 128; s > 0; s >>= 1) {
    if (t < s) red[t] += red[t + s];
    __syncthreads();
  }
  if (t == 0) out[B * L] = red[0] / (float)(B * L);
}

// ---------------------------------------------------------------------------
extern "C" void kernel_launch(void* const* d_in, const int* in_sizes, int n_in,
                              void* d_out, int out_size, void* d_ws,
                              size_t ws_size, hipStream_t stream) {
  using namespace cfg;
  (void)in_sizes; (void)n_in; (void)out_size; (void)ws_size;
  const float* codes = (const float*)d_in[0];
  const int* y = (const int*)d_in[1];
  const float* fc_w = (const float*)d_in[2];
  const float* fc_b = (const float*)d_in[3];
  const float* in_w0 = (const float*)d_in[4];
  const float* in_b0 = (const float*)d_in[5];
  const float* in_w1 = (const float*)d_in[6];
  const float* in_b1 = (const float*)d_in[7];
  const float* in_ws = (const float*)d_in[8];
  const float* leak = (const float*)d_in[9];
  const float* hyper_w = (const float*)d_in[10];
  const float* hyper_b = (const float*)d_in[11];
  const float* out_w0 = (const float*)d_in[12];
  const float* out_b0 = (const float*)d_in[13];
  const float* out_w1 = (const float*)d_in[14];
  const float* out_b1 = (const float*)d_in[15];
  const float* l1_w0 = (const float*)d_in[16];
  const float* l1_b0 = (const float*)d_in[17];
  const float* l1_w1 = (const float*)d_in[18];
  const float* l1_b1 = (const float*)d_in[19];
  const float* l2_w0 = (const float*)d_in[20];
  const float* l2_b0 = (const float*)d_in[21];
  const float* l2_w1 = (const float*)d_in[22];
  const float* l2_b1 = (const float*)d_in[23];
  const float* l2_ws = (const float*)d_in[24];

  float* wsf = (float*)d_ws;
  float* yemb = wsf;                               // B*EMB
  float* p = yemb + (size_t)B * EMB;               // B*P
  float* x = p + (size_t)B * P;                    // B*C*L
  float* lossp = x + (size_t)B * C * L;            // B
  float* outf = (float*)d_out;

  yemb_k<<<dim3(B), dim3(256), 0, stream>>>(fc_w, fc_b, y, yemb);
  hyper_k<<<dim3(579, 16), dim3(128), 0, stream>>>(yemb, hyper_w, hyper_b, p);
  pre_k<<<dim3(B), dim3(256), 0, stream>>>(codes, in_w0, in_b0, in_w1, in_b1,
                                           in_ws, x);
  iter_k<<<dim3(B), dim3(1024), 0, stream>>>(p, leak, x);
  post_k<<<dim3(B), dim3(256), 0, stream>>>(x, codes, out_w0, out_b0, out_w1,
                                            out_b1, l1_w0, l1_b0, l1_w1, l1_b1,
                                            l2_w0, l2_b0, l2_w1, l2_b1, l2_ws,
                                            outf, lossp);
  loss_k<<<dim3(1), dim3(256), 0, stream>>>(lossp, outf);
}